// AdMSoftmaxLoss_39797166965247
// MI455X (gfx1250) — compile-verified
//
#include <hip/hip_runtime.h>

// ---------------------------------------------------------------------------
// AdMSoftmax loss, MI455X (gfx1250, wave32).
// Memory-bound: W (204.8MB fp32) streamed once from HBM (~8.8us @ 23.3TB/s).
// fp32 WMMA 16x16x4 keeps full precision; matrix compute (~105 GFLOP) is far
// below the fp32 matrix ceiling, so the job is: stream W once, keep the
// per-tile epilogue (exp + 16-lane reduce) off the LDS pipe (DPP16 butterfly,
// raw v_exp_f32), and stage W tiles via the async global->LDS path.
// ---------------------------------------------------------------------------

typedef __attribute__((ext_vector_type(2))) float v2f;
typedef __attribute__((ext_vector_type(8))) float v8f;
typedef __attribute__((ext_vector_type(4))) int   v4i;

#define EMB   512
#define NCLS  100000
#define NROW  1024
#define SCALE 30.0f
#define MARG  0.4f
#define CT    64                 // classes per workgroup
#define LDSW_STRIDE 516          // 512 + 4 float pad -> conflict-free ds_load_b64
#define NTILE ((NCLS + CT - 1) / CT)   // 1563
#define S_LOG2E 43.28085122666891f     // 30 * log2(e)

#if __has_builtin(__builtin_amdgcn_global_load_async_to_lds_b128) && \
    __has_builtin(__builtin_amdgcn_s_wait_asynccnt)
#define USE_ASYNC_LDS 1
#endif

#define GLOBAL_AS __attribute__((address_space(1)))
#define LDS_AS    __attribute__((address_space(3)))

// DPP16 butterfly add: masks {1,2,7,15} span the 16-lane group.
// 0xB1 = quad_perm[1,0,3,2] (xor1), 0x4E = quad_perm[2,3,0,1] (xor2),
// 0x141 = row_half_mirror (xor7), 0x140 = row_mirror (xor15).
template <int CTRL>
__device__ __forceinline__ float dppAdd(float v) {
    int t = __builtin_amdgcn_update_dpp(0, __float_as_int(v), CTRL, 0xF, 0xF, true);
    return v + __int_as_float(t);
}

// ---------------- kernel 1: row-normalize x -> xn --------------------------
__global__ void k_normalize(const float* __restrict__ x, float* __restrict__ xn) {
    int wave = threadIdx.x >> 5;
    int lane = threadIdx.x & 31;
    int row  = blockIdx.x * 8 + wave;          // grid 128 x 256 threads
    const float* xr = x + (size_t)row * EMB;
    float ss = 0.f;
    for (int e = lane; e < EMB; e += 32) { float v = xr[e]; ss += v * v; }
    for (int m = 16; m; m >>= 1) ss += __shfl_xor(ss, m, 32);
    float inv = 1.0f / sqrtf(ss);
    float* xo = xn + (size_t)row * EMB;
    for (int e = lane; e < EMB; e += 32) xo[e] = xr[e] * inv;
}

// ---------------- kernel 2: fused GEMM + exp-sum per 64-class tile ---------
// One WG = 64 classes for ALL 1024 rows; W tile staged in LDS once (async
// global->LDS when available). 8 waves = 4 class-subtiles x 2 row-parities.
// Per 16x16 tile: 128 fp32 WMMAs over K=512, then raw v_exp_f32 + DPP16
// butterfly reduce -> per-row partial exp-sums (deterministic, no atomics).
__global__ void k_gemm_expsum(const float* __restrict__ xn,
                              const float* __restrict__ W,
                              float* __restrict__ partial) {
    extern __shared__ float smem[];
    float* ldsW = smem;                      // CT * LDSW_STRIDE floats (132KB)
    float* lp   = smem + CT * LDSW_STRIDE;   // 4 * NROW floats (16KB)

    const int c0    = blockIdx.x * CT;
    const int valid = (NCLS - c0 < CT) ? (NCLS - c0) : CT;   // 1..64

    // zero-fill pad class rows (only last WG): logits become exactly 0 ->
    // exp contributes exactly 1.0, subtracted exactly at the end.
    for (int i = threadIdx.x; i < (CT - valid) * (EMB / 4); i += blockDim.x) {
        int r  = valid + (i >> 7);
        int c4 = i & 127;
        *(float4*)(ldsW + r * LDSW_STRIDE + c4 * 4) = make_float4(0.f, 0.f, 0.f, 0.f);
    }

#ifdef USE_ASYNC_LDS
    for (int i = threadIdx.x; i < valid * (EMB / 4); i += blockDim.x) {
        int r  = i >> 7;               // EMB/4 = 128 float4 per row
        int c4 = i & 127;
        GLOBAL_AS v4i* gsrc =
            (GLOBAL_AS v4i*)(W + (size_t)(c0 + r) * EMB + (size_t)c4 * 4);
        LDS_AS v4i* ldst = (LDS_AS v4i*)(ldsW + r * LDSW_STRIDE + c4 * 4);
        __builtin_amdgcn_global_load_async_to_lds_b128(gsrc, ldst, 0, 0);
    }
    __builtin_amdgcn_s_wait_asynccnt(0);
#else
    for (int i = threadIdx.x; i < valid * (EMB / 4); i += blockDim.x) {
        int r  = i >> 7;
        int c4 = i & 127;
        float4 v = *(const float4*)(W + (size_t)(c0 + r) * EMB + (size_t)c4 * 4);
        *(float4*)(ldsW + r * LDSW_STRIDE + c4 * 4) = v;
    }
#endif
    __syncthreads();

    const int wave = threadIdx.x >> 5;
    const int lane = threadIdx.x & 31;
    const int csub = wave & 3;                 // class sub-tile 0..3 (16 cls)
    const int rpar = wave >> 2;                // row-tile parity 0..1
    const int col  = lane & 15;                // M for A, N for B
    const int koff = (lane >> 4) << 1;         // K pair select per half-wave
    const int hi8  = (lane >> 4) << 3;         // upper half owns rows M+8..15

    const float* bbase = ldsW + (csub * 16 + col) * LDSW_STRIDE + koff;

    for (int rt = rpar; rt < NROW / 16; rt += 2) {
        const int m0 = rt * 16;
        const float* abase = xn + (size_t)(m0 + col) * EMB + koff;

        v8f c = {0.f, 0.f, 0.f, 0.f, 0.f, 0.f, 0.f, 0.f};
#pragma unroll 8
        for (int k = 0; k < EMB; k += 4) {
            v2f a = *(const v2f*)(abase + k);      // global, L2-resident xn
            v2f b = *(const v2f*)(bbase + k);      // LDS, conflict-free
            c = __builtin_amdgcn_wmma_f32_16x16x4_f32(
                    /*neg_a=*/false, a, /*neg_b=*/false, b,
                    /*c_mod=*/(short)0, c, /*reuse_a=*/false, /*reuse_b=*/false);
        }

        // exp(S*logit) = 2^(logit*S*log2e); arguments bounded (~|26|), so raw
        // v_exp_f32 is safe. Reduce over 16 columns with pure-VALU DPP16.
#pragma unroll
        for (int r = 0; r < 8; ++r) {
            float e = __builtin_amdgcn_exp2f(c[r] * S_LOG2E);
            e = dppAdd<0xB1>(e);    // xor 1
            e = dppAdd<0x4E>(e);    // xor 2
            e = dppAdd<0x141>(e);   // xor 7 (row_half_mirror)
            e = dppAdd<0x140>(e);   // xor 15 (row_mirror)
            if (col == 0)
                lp[csub * NROW + m0 + hi8 + r] = e;   // written exactly once
        }
    }
    __syncthreads();

    // combine the 4 class-subtiles, subtract exact pad contribution (1.0 each)
    const float pad = (float)(CT - valid);
    for (int row = threadIdx.x; row < NROW; row += blockDim.x) {
        float p = lp[row] + lp[NROW + row] + lp[2 * NROW + row] + lp[3 * NROW + row];
        partial[(size_t)blockIdx.x * NROW + row] = p - pad;
    }
}

// ---------------- kernel 3: per-row loss -----------------------------------
__global__ void k_rowloss(const float* __restrict__ xn,
                          const float* __restrict__ W,
                          const int* __restrict__ labels,
                          const float* __restrict__ partial,
                          float* __restrict__ Lrow) {
    int wave = threadIdx.x >> 5;
    int lane = threadIdx.x & 31;
    int row  = blockIdx.x * 8 + wave;          // grid 128 x 256 threads

    float s = 0.f;                              // sum_j exp(S*wf_j)
    for (int t = lane; t < NTILE; t += 32)
        s += partial[(size_t)t * NROW + row];

    int lab = labels[row];
    const float* xr = xn + (size_t)row * EMB;
    const float* wr = W + (size_t)lab * EMB;
    float d = 0.f;                              // wf_y
    for (int e = lane; e < EMB; e += 32) d += xr[e] * wr[e];

    for (int m = 16; m; m >>= 1) {
        s += __shfl_xor(s, m, 32);
        d += __shfl_xor(d, m, 32);
    }
    if (lane == 0) {
        float numer = SCALE * (d - MARG);
        float denom = expf(numer) + s - expf(SCALE * d);
        Lrow[row] = numer - logf(denom);
    }
}

// ---------------- kernel 4: -mean ------------------------------------------
__global__ void k_finalize(const float* __restrict__ Lrow, float* __restrict__ out) {
    __shared__ float sm[8];
    int lane = threadIdx.x & 31;
    int wave = threadIdx.x >> 5;
    float s = 0.f;
    for (int i = threadIdx.x; i < NROW; i += 256) s += Lrow[i];
    for (int m = 16; m; m >>= 1) s += __shfl_xor(s, m, 32);
    if (lane == 0) sm[wave] = s;
    __syncthreads();
    if (threadIdx.x == 0) {
        float t = 0.f;
        for (int w = 0; w < 8; ++w) t += sm[w];
        out[0] = -t / (float)NROW;
    }
}

// ---------------------------------------------------------------------------
extern "C" void kernel_launch(void* const* d_in, const int* in_sizes, int n_in,
                              void* d_out, int out_size, void* d_ws, size_t ws_size,
                              hipStream_t stream) {
    const float* x      = (const float*)d_in[0];   // (1024, 512) fp32
    const int*   labels = (const int*)d_in[1];     // (1024,) int32
    const float* W      = (const float*)d_in[2];   // (100000, 512) fp32
    float* out = (float*)d_out;

    // workspace layout: xn | partial | Lrow  (~8.4 MB)
    float* xn      = (float*)d_ws;
    float* partial = xn + (size_t)NROW * EMB;
    float* Lrow    = partial + (size_t)NTILE * NROW;

    k_normalize<<<128, 256, 0, stream>>>(x, xn);

    size_t shbytes = (size_t)(CT * LDSW_STRIDE + 4 * NROW) * sizeof(float); // ~148.5KB
    k_gemm_expsum<<<NTILE, 256, shbytes, stream>>>(xn, W, partial);

    k_rowloss<<<128, 256, 0, stream>>>(xn, W, labels, partial, Lrow);
    k_finalize<<<1, 256, 0, stream>>>(Lrow, out);
}